// GraphAttentionConvLayer_12077448037163
// MI455X (gfx1250) — compile-verified
//
#include <hip/hip_runtime.h>

#define B_ 8
#define N_ 4096
#define S_ 1024
#define K_ 32
#define R2_ 0.0225f
#define ALPHA_ 0.2f
#define EPS_ 1e-5f
#define LDH 131   // LDS row stride for 131-wide attention input / 128-wide activations

typedef float v2f __attribute__((ext_vector_type(2)));
typedef float v8f __attribute__((ext_vector_type(8)));

// ---------------------------------------------------------------------------
// Kernel 0: fold BatchNorm (eval mode) into conv weights; zero-pad K dims.
// ---------------------------------------------------------------------------
__global__ void prep_kernel(
    const float* __restrict__ w0, const float* __restrict__ b0,
    const float* __restrict__ g0, const float* __restrict__ be0,
    const float* __restrict__ m0, const float* __restrict__ v0,
    const float* __restrict__ w1, const float* __restrict__ b1,
    const float* __restrict__ g1, const float* __restrict__ be1,
    const float* __restrict__ m1, const float* __restrict__ v1,
    const float* __restrict__ w2, const float* __restrict__ b2,
    const float* __restrict__ g2, const float* __restrict__ be2,
    const float* __restrict__ m2, const float* __restrict__ v2,
    const float* __restrict__ a,
    float* __restrict__ W0p, float* __restrict__ b0p,
    float* __restrict__ W1p, float* __restrict__ b1p,
    float* __restrict__ W2p, float* __restrict__ b2p,
    float* __restrict__ ap)
{
  int t = threadIdx.x, nt = blockDim.x;
  // layer 0: [64 x 9] -> padded [64 x 12]
  for (int i = t; i < 64 * 12; i += nt) {
    int o = i / 12, c = i - o * 12;
    float s = g0[o] * rsqrtf(v0[o] + EPS_);
    W0p[i] = (c < 9) ? s * w0[o * 9 + c] : 0.f;
  }
  for (int o = t; o < 64; o += nt) {
    float s = g0[o] * rsqrtf(v0[o] + EPS_);
    b0p[o] = s * (b0[o] - m0[o]) + be0[o];
  }
  // layer 1: [64 x 64]
  for (int i = t; i < 64 * 64; i += nt) {
    int o = i / 64;
    float s = g1[o] * rsqrtf(v1[o] + EPS_);
    W1p[i] = s * w1[i];
  }
  for (int o = t; o < 64; o += nt) {
    float s = g1[o] * rsqrtf(v1[o] + EPS_);
    b1p[o] = s * (b1[o] - m1[o]) + be1[o];
  }
  // layer 2: [128 x 64]
  for (int i = t; i < 128 * 64; i += nt) {
    int o = i / 64;
    float s = g2[o] * rsqrtf(v2[o] + EPS_);
    W2p[i] = s * w2[i];
  }
  for (int o = t; o < 128; o += nt) {
    float s = g2[o] * rsqrtf(v2[o] + EPS_);
    b2p[o] = s * (b2[o] - m2[o]) + be2[o];
  }
  // attention matrix: [131 x 128] -> padded [132 x 128] (row 131 = zeros)
  for (int i = t; i < 132 * 128; i += nt) {
    int r = i / 128;
    ap[i] = (r < 131) ? a[i] : 0.f;
  }
}

// ---------------------------------------------------------------------------
// Kernel 1: farthest point sampling. One block (1024 threads) per batch.
// Each thread owns 4 points in registers; argmax via wave shfl + LDS.
// ---------------------------------------------------------------------------
__global__ __launch_bounds__(1024) void fps_kernel(
    const float* __restrict__ xyz, int* __restrict__ fpsIdx,
    float* __restrict__ newXyz, float* __restrict__ outXyz)
{
  int b = blockIdx.x, t = threadIdx.x;
  const float* xb = xyz + b * 3 * N_;
  float px[4], py[4], pz[4], dist[4];
#pragma unroll
  for (int j = 0; j < 4; ++j) {
    int n = t + j * 1024;
    px[j] = xb[n]; py[j] = xb[N_ + n]; pz[j] = xb[2 * N_ + n];
    dist[j] = 1e10f;
  }
  __shared__ float sC[3];
  __shared__ int sFar;
  __shared__ float wv[32];
  __shared__ int wi[32];
  if (t == 0) sFar = 0;
  __syncthreads();

  for (int i = 0; i < S_; ++i) {
    if (t == 0) {
      int f = sFar;
      fpsIdx[b * S_ + i] = f;
      float cx = xb[f], cy = xb[N_ + f], cz = xb[2 * N_ + f];
      newXyz[(b * S_ + i) * 3 + 0] = cx;
      newXyz[(b * S_ + i) * 3 + 1] = cy;
      newXyz[(b * S_ + i) * 3 + 2] = cz;
      outXyz[b * 3 * S_ + 0 * S_ + i] = cx;   // new_xyz.T  [B,3,S]
      outXyz[b * 3 * S_ + 1 * S_ + i] = cy;
      outXyz[b * 3 * S_ + 2 * S_ + i] = cz;
      sC[0] = cx; sC[1] = cy; sC[2] = cz;
    }
    __syncthreads();
    float cx = sC[0], cy = sC[1], cz = sC[2];
    float bv = -1.f; int bi = 0;
#pragma unroll
    for (int j = 0; j < 4; ++j) {
      float dx = px[j] - cx, dy = py[j] - cy, dz = pz[j] - cz;
      float dd = dx * dx + dy * dy + dz * dz;
      dist[j] = fminf(dist[j], dd);
      if (dist[j] > bv) { bv = dist[j]; bi = t + j * 1024; }
    }
    // wave32 reduction, tie-break to lowest index (matches jnp.argmax)
    for (int off = 16; off; off >>= 1) {
      float ov = __shfl_xor(bv, off, 32);
      int oi = __shfl_xor(bi, off, 32);
      if (ov > bv || (ov == bv && oi < bi)) { bv = ov; bi = oi; }
    }
    int lane = t & 31, wid = t >> 5;
    if (lane == 0) { wv[wid] = bv; wi[wid] = bi; }
    __syncthreads();
    if (wid == 0) {
      bv = wv[lane]; bi = wi[lane];
      for (int off = 16; off; off >>= 1) {
        float ov = __shfl_xor(bv, off, 32);
        int oi = __shfl_xor(bi, off, 32);
        if (ov > bv || (ov == bv && oi < bi)) { bv = ov; bi = oi; }
      }
      if (lane == 0) sFar = bi;
    }
    __syncthreads();
  }
}

// ---------------------------------------------------------------------------
// Kernel 2: ball query. One wave32 per (b,s); in-order first-K selection via
// ballot + prefix popcount (identical to sort-indices-take-K semantics).
// ---------------------------------------------------------------------------
__global__ __launch_bounds__(256) void ball_kernel(
    const float* __restrict__ xyz, const float* __restrict__ newXyz,
    int* __restrict__ ballIdx)
{
  int lane = threadIdx.x & 31;
  int bs = blockIdx.x * 8 + (threadIdx.x >> 5);
  int b = bs >> 10;
  const float* xb = xyz + b * 3 * N_;
  float cx = newXyz[bs * 3 + 0], cy = newXyz[bs * 3 + 1], cz = newXyz[bs * 3 + 2];
  int cnt = 0, firstIdx = 0;
  for (int c = 0; c < N_ / 32 && cnt < K_; ++c) {
    int n = c * 32 + lane;
    float dx = xb[n] - cx, dy = xb[N_ + n] - cy, dz = xb[2 * N_ + n] - cz;
    bool hit = (dx * dx + dy * dy + dz * dz) <= R2_;
    unsigned mask = __builtin_amdgcn_ballot_w32(hit);
    if (mask) {
      if (cnt == 0) firstIdx = c * 32 + (__ffs(mask) - 1);
      int pos = cnt + __popc(mask & ((1u << lane) - 1u));
      if (hit && pos < K_) ballIdx[bs * K_ + pos] = n;
      cnt += __popc(mask);
    }
  }
  if (cnt < K_ && lane >= cnt && lane < K_) ballIdx[bs * K_ + lane] = firstIdx;
}

// ---------------------------------------------------------------------------
// fp32 WMMA GEMM helper:  Xout[M][N] = relu( Xin[M][K] * W^T + bias )
// W is row-major [Nout][Kdim] (B[k][n] = W[n][k]).
// ISA f32 16x16x4 layouts: A lane = M (lo), lane-half selects K pair;
// B symmetric; C/D: vgpr j -> M = j + 8*hi, N = lo.
// `wave` must be wave-uniform (pass through readfirstlane) so the tile loop
// is scalar-controlled and EXEC stays all-ones around every WMMA.
// ---------------------------------------------------------------------------
static __device__ __forceinline__ void gemm_wmma(
    const float* __restrict__ Xin, int ldin,
    const float* __restrict__ W, int kdim, int Ktiles,
    const float* __restrict__ bias,
    float* __restrict__ Xout, int ldout,
    int Mtiles, int Ntiles, int wave, int lane)
{
  int lo = lane & 15, hi = lane >> 4;
  int total = Mtiles * Ntiles;
  for (int tt = wave; tt < total; tt += 4) {
    int mt = tt / Ntiles, nt = tt - mt * Ntiles;
    int n = nt * 16 + lo;
    float bb = bias[n];
    v8f acc = {bb, bb, bb, bb, bb, bb, bb, bb};
    const float* arow = Xin + (mt * 16 + lo) * ldin + 2 * hi;
    const float* brow = W + n * kdim + 2 * hi;
    for (int kt = 0; kt < Ktiles; ++kt) {
      v2f af, bf;
      af.x = arow[kt * 4]; af.y = arow[kt * 4 + 1];
      bf.x = brow[kt * 4]; bf.y = brow[kt * 4 + 1];
      acc = __builtin_amdgcn_wmma_f32_16x16x4_f32(
          false, af, false, bf, (short)0, acc, false, false);
    }
    int mbase = mt * 16 + hi * 8;
#pragma unroll
    for (int j = 0; j < 8; ++j)
      Xout[(mbase + j) * ldout + n] = fmaxf(acc[j], 0.f);
  }
}

// ---------------------------------------------------------------------------
// Kernel 3: fused gather + MLP (3 layers, WMMA) + graph attention.
// One block (4 wave32s) per (b,s) sample; 33 live rows padded to 48 (3 M-tiles).
// ---------------------------------------------------------------------------
__global__ __launch_bounds__(128) void fused_kernel(
    const float* __restrict__ xyz, const float* __restrict__ pts,
    const int* __restrict__ fpsIdx, const int* __restrict__ ballIdx,
    const float* __restrict__ newXyz,
    const float* __restrict__ W0p, const float* __restrict__ b0p,
    const float* __restrict__ W1p, const float* __restrict__ b1p,
    const float* __restrict__ W2p, const float* __restrict__ b2p,
    const float* __restrict__ ap,
    float* __restrict__ outFeat)
{
  __shared__ float x0[48 * 13];      // layer-0 input, K padded 9->12
  __shared__ float bufA[48 * LDH];   // activations ping
  __shared__ float bufB[48 * LDH];   // activations pong / G matrix / e matrix
  __shared__ int sIdx[32];
  __shared__ float sC[3];
  __shared__ int sFps;

  int t = threadIdx.x;
  int bs = blockIdx.x;
  int b = bs >> 10, s = bs & 1023;
  int lane = t & 31;
  // wave id is uniform across the wave: force scalar so tile loops use SALU
  // control flow (no EXEC masking around v_wmma).
  int wave = __builtin_amdgcn_readfirstlane(t >> 5);

  for (int i = t; i < 48 * 13; i += 128) x0[i] = 0.f;
  if (t < 32) sIdx[t] = ballIdx[bs * K_ + t];
  if (t == 0) {
    sC[0] = newXyz[bs * 3 + 0];
    sC[1] = newXyz[bs * 3 + 1];
    sC[2] = newXyz[bs * 3 + 2];
    sFps = fpsIdx[bs];
  }
  __syncthreads();

  const float* xb = xyz + b * 3 * N_;
  const float* pb = pts + b * 6 * N_;
  if (t < 32) {                       // neighbor rows: [rel_xyz, grouped_pts]
    int n = sIdx[t];
    x0[t * 13 + 0] = xb[n] - sC[0];
    x0[t * 13 + 1] = xb[N_ + n] - sC[1];
    x0[t * 13 + 2] = xb[2 * N_ + n] - sC[2];
#pragma unroll
    for (int d = 0; d < 6; ++d) x0[t * 13 + 3 + d] = pb[d * N_ + n];
  } else if (t == 32) {               // center row: [new_xyz, fps_pts]
    int n = sFps;
    x0[32 * 13 + 0] = sC[0];
    x0[32 * 13 + 1] = sC[1];
    x0[32 * 13 + 2] = sC[2];
    for (int d = 0; d < 6; ++d) x0[32 * 13 + 3 + d] = pb[d * N_ + n];
  }
  __syncthreads();

  gemm_wmma(x0, 13, W0p, 12, 3, b0p, bufA, LDH, 3, 4, wave, lane);     // 9->64
  __syncthreads();
  gemm_wmma(bufA, LDH, W1p, 64, 16, b1p, bufB, LDH, 3, 4, wave, lane); // 64->64
  __syncthreads();
  gemm_wmma(bufB, LDH, W2p, 64, 16, b2p, bufA, LDH, 3, 8, wave, lane); // 64->128
  __syncthreads();

  // Build G = [dp | dh] in bufB rows 0..31, cols 0..130
  {
    int k = t & 31;
    int part = t >> 5;
    if (part == 0) {
      bufB[k * LDH + 0] = -x0[k * 13 + 0];  // dp = new_xyz - grouped = -rel
      bufB[k * LDH + 1] = -x0[k * 13 + 1];
      bufB[k * LDH + 2] = -x0[k * 13 + 2];
    }
    for (int j = part; j < 128; j += 4)
      bufB[k * LDH + 3 + j] = bufA[32 * LDH + j] - bufA[k * LDH + j];
  }
  __syncthreads();

  // e = leaky_relu(G @ a): M=32 (2 tiles), N=128 (8 tiles), K=131 (33 tiles).
  // Accumulators held in registers across the barrier so e can overwrite G.
  int lo = lane & 15, hi = lane >> 4;
  v8f accs[4];
#pragma unroll
  for (int i = 0; i < 4; ++i) {
    int tt = wave + i * 4;
    int mt = tt >> 3, nt = tt & 7;
    int n = nt * 16 + lo;
    v8f acc = {0.f, 0.f, 0.f, 0.f, 0.f, 0.f, 0.f, 0.f};
    const float* arow = bufB + (mt * 16 + lo) * LDH + 2 * hi;
    for (int kt = 0; kt < 33; ++kt) {
      v2f af, bf;
      af.x = arow[kt * 4]; af.y = arow[kt * 4 + 1];
      int k0 = kt * 4 + 2 * hi;
      bf.x = ap[k0 * 128 + n];
      bf.y = ap[(k0 + 1) * 128 + n];   // row 131 is zero-padded
      acc = __builtin_amdgcn_wmma_f32_16x16x4_f32(
          false, af, false, bf, (short)0, acc, false, false);
    }
    accs[i] = acc;
  }
  __syncthreads();
#pragma unroll
  for (int i = 0; i < 4; ++i) {
    int tt = wave + i * 4;
    int mt = tt >> 3, nt = tt & 7;
    int n = nt * 16 + lo;
    int mbase = mt * 16 + hi * 8;
#pragma unroll
    for (int j = 0; j < 8; ++j) {
      float v = accs[i][j];
      bufB[(mbase + j) * LDH + n] = (v > 0.f) ? v : ALPHA_ * v;
    }
  }
  __syncthreads();

  // Softmax over K per channel, weighted sum of neighbor features.
  {
    int o = t;  // 0..127
    float mx = -1e30f;
    for (int k = 0; k < K_; ++k) mx = fmaxf(mx, bufB[k * LDH + o]);
    float Z = 0.f, acc = 0.f;
    for (int k = 0; k < K_; ++k) {
      float w = expf(bufB[k * LDH + o] - mx);
      Z += w;
      acc += w * bufA[k * LDH + o];
    }
    outFeat[(b * 128 + o) * S_ + s] = acc / Z;   // out.T  [B,128,S]
  }
}

// ---------------------------------------------------------------------------
extern "C" void kernel_launch(void* const* d_in, const int* in_sizes, int n_in,
                              void* d_out, int out_size, void* d_ws, size_t ws_size,
                              hipStream_t stream)
{
  (void)in_sizes; (void)n_in; (void)out_size; (void)ws_size;
  const float* xyz = (const float*)d_in[0];
  const float* pts = (const float*)d_in[1];
  const float* w0 = (const float*)d_in[2];
  const float* b0 = (const float*)d_in[3];
  const float* g0 = (const float*)d_in[4];
  const float* be0 = (const float*)d_in[5];
  const float* m0 = (const float*)d_in[6];
  const float* v0 = (const float*)d_in[7];
  const float* w1 = (const float*)d_in[8];
  const float* b1 = (const float*)d_in[9];
  const float* g1 = (const float*)d_in[10];
  const float* be1 = (const float*)d_in[11];
  const float* m1 = (const float*)d_in[12];
  const float* v1 = (const float*)d_in[13];
  const float* w2 = (const float*)d_in[14];
  const float* b2 = (const float*)d_in[15];
  const float* g2 = (const float*)d_in[16];
  const float* be2 = (const float*)d_in[17];
  const float* m2 = (const float*)d_in[18];
  const float* v2 = (const float*)d_in[19];
  const float* a = (const float*)d_in[20];
  float* out = (float*)d_out;

  char* w = (char*)d_ws;
  int* fpsIdx = (int*)w;   w += (size_t)B_ * S_ * sizeof(int);
  int* ballIdx = (int*)w;  w += (size_t)B_ * S_ * K_ * sizeof(int);
  float* newXyz = (float*)w; w += (size_t)B_ * S_ * 3 * sizeof(float);
  float* W0p = (float*)w;  w += 64 * 12 * sizeof(float);
  float* b0p = (float*)w;  w += 64 * sizeof(float);
  float* W1p = (float*)w;  w += 64 * 64 * sizeof(float);
  float* b1p = (float*)w;  w += 64 * sizeof(float);
  float* W2p = (float*)w;  w += 128 * 64 * sizeof(float);
  float* b2p = (float*)w;  w += 128 * sizeof(float);
  float* ap = (float*)w;   w += 132 * 128 * sizeof(float);

  prep_kernel<<<1, 256, 0, stream>>>(w0, b0, g0, be0, m0, v0,
                                     w1, b1, g1, be1, m1, v1,
                                     w2, b2, g2, be2, m2, v2, a,
                                     W0p, b0p, W1p, b1p, W2p, b2p, ap);
  fps_kernel<<<B_, 1024, 0, stream>>>(xyz, fpsIdx, newXyz, out);
  ball_kernel<<<B_ * S_ / 8, 256, 0, stream>>>(xyz, newXyz, ballIdx);
  fused_kernel<<<B_ * S_, 128, 0, stream>>>(xyz, pts, fpsIdx, ballIdx, newXyz,
                                            W0p, b0p, W1p, b1p, W2p, b2p, ap,
                                            out + (size_t)B_ * 3 * S_);
}